// TestNet_89438398972397
// MI455X (gfx1250) — compile-verified
//
#include <hip/hip_runtime.h>
#include <math.h>

typedef __attribute__((ext_vector_type(2))) float v2f;
typedef __attribute__((ext_vector_type(8))) float v8f;

#define TOTAL 16777216   // 8*512*64*64
#define PCH   4624       // padded channel plane: 68*68
#define PB    591872     // padded batch: 128*68*68

// ---------------------------------------------------------------------------
// DCT basis generation (orthonormal DCT-II), double-precision trig once.
// ---------------------------------------------------------------------------
__global__ void gen_dct(float* Ac, float* Ak) {
    const int i = blockIdx.x * blockDim.x + threadIdx.x;
    const double PI = 3.14159265358979323846;
    if (i < 4096) {
        int k = i >> 6, n = i & 63;
        double v = cos(PI / 64.0 * (double)k * ((double)n + 0.5)) * sqrt(2.0 / 64.0);
        if (k == 0) v *= 0.7071067811865476;
        Ac[i] = (float)v;
    } else if (i < 4096 + 262144) {
        int j = i - 4096;
        int k = j >> 9, c = j & 511;
        double v = cos(PI / 512.0 * (double)k * ((double)c + 0.5)) * sqrt(2.0 / 512.0);
        if (k == 0) v *= 0.7071067811865476;
        Ak[j] = (float)v;
    }
}

__global__ void zero_buf(float* p, long long n) {
    long long i = (long long)blockIdx.x * blockDim.x + threadIdx.x;
    const long long stride = (long long)gridDim.x * blockDim.x;
    for (; i < n; i += stride) p[i] = 0.0f;
}

// ---------------------------------------------------------------------------
// Strided WMMA fp32 GEMM, all strides/K compile-time:
//   O[i,j] = sum_k L(i,k) * R(k,j)
// L(i,k) = Lb[z*lbatch + i*LS0 + k*LS1]  (same pattern for R, O).
// NT = number of 16-wide N subtiles per wave (4 -> 64-wide, 8 -> 128-wide).
// Block = 128 threads = 4 waves; block tile 64 x (NT*16).
// One base pointer per operand; subtile/K offsets are compile-time imms.
// OPAD: write O into zero-padded (.,68,68) layout (for conv input halo).
// ---------------------------------------------------------------------------
template<int NT, long long LS0, long long LS1, long long RS0, long long RS1,
         long long OS0, long long OS1, int KDIM,
         bool MASK, bool HASBIAS, bool RELU, bool OPAD>
__global__ void __launch_bounds__(128)
wmma_gemm(const float* __restrict__ Lb, long long lbatch,
          const float* __restrict__ Rb, long long rbatch,
          float* __restrict__ Ob, long long obatch,
          const float* __restrict__ thrp, const float* __restrict__ bias)
{
    const int lane  = threadIdx.x & 31;
    const int wave  = threadIdx.x >> 5;
    const int mrow  = lane & 15;     // row within A tile / col within B,C,D tiles
    const int khalf = lane >> 4;     // lane-half selects K pair (A,B) / M+8 (C,D)
    const int m0 = blockIdx.x * 64 + wave * 16;
    const int n0 = blockIdx.y * (NT * 16);

    const float* L = Lb + (long long)blockIdx.z * lbatch;
    const float* R = Rb + (long long)blockIdx.z * rbatch;
    float*       O = Ob + (long long)blockIdx.z * obatch;

    const float thrv = MASK ? *thrp : 0.0f;

    const float* lp = L + (long long)(m0 + mrow) * LS0 + (long long)(2 * khalf) * LS1;
    const float* rb = R + (long long)(2 * khalf) * RS0 + (long long)(n0 + mrow) * RS1;

    v8f acc[NT] = {};
#pragma unroll 4
    for (int k0 = 0; k0 < KDIM; k0 += 4) {
        v2f a;
        a.x = lp[0];
        a.y = lp[LS1];
        lp += 4 * LS1;
#pragma unroll
        for (int t = 0; t < NT; ++t) {
            float r0 = rb[t * 16 * RS1];
            float r1 = rb[t * 16 * RS1 + RS0];
            if (MASK) {
                r0 = (fabsf(r0) >= thrv) ? r0 : 0.0f;
                r1 = (fabsf(r1) >= thrv) ? r1 : 0.0f;
            }
            v2f bf; bf.x = r0; bf.y = r1;
            acc[t] = __builtin_amdgcn_wmma_f32_16x16x4_f32(
                false, a, false, bf, (short)0, acc[t], false, false);
        }
        rb += 4 * RS0;
    }
    // Epilogue. C/D layout: VGPR r -> M = r (lanes 0-15), M = r+8 (lanes 16-31)
#pragma unroll
    for (int t = 0; t < NT; ++t) {
        const int n = n0 + t * 16 + mrow;
#pragma unroll
        for (int r = 0; r < 8; ++r) {
            const int m = m0 + r + 8 * khalf;
            float v = acc[t][r];
            if (HASBIAS) v += bias[m];
            if (RELU) v = fmaxf(v, 0.0f);
            if (OPAD) {
                const int yy = n >> 6, xx = n & 63;
                O[(long long)m * PCH + (long long)(yy + 2) * 68 + xx + 2] = v;
            } else {
                O[(long long)m * OS0 + (long long)n * OS1] = v;
            }
        }
    }
}

// ---------------------------------------------------------------------------
// 3x3 dilation-2 conv (pad 2), 128->128 channels, 64x64 spatial, as
// implicit-GEMM WMMA over a zero-padded (128,68,68) input: every load is
// in-bounds (halo pre-zeroed), so the inner loop is pure clause-able loads
// at immediate offsets + v_wmma. Fused bias + ReLU. Grid (2, 64, 8).
// ---------------------------------------------------------------------------
__global__ void __launch_bounds__(128)
conv3x3_dil2_wmma(const float* __restrict__ in, const float* __restrict__ W,
                  const float* __restrict__ bias, float* __restrict__ out)
{
    const int lane  = threadIdx.x & 31;
    const int wave  = threadIdx.x >> 5;
    const int mrow  = lane & 15;
    const int khalf = lane >> 4;
    const int m0 = blockIdx.x * 64 + wave * 16;   // out-channel base
    const int n0 = blockIdx.y * 64;               // flattened spatial base
    const int y  = n0 >> 6;                       // output row (block-uniform)
    const float* inb  = in  + (long long)blockIdx.z * PB;
    float*       outb = out + (long long)blockIdx.z * 524288;

    v8f acc[4] = {};
#pragma unroll
    for (int ky = 0; ky < 3; ++ky) {
#pragma unroll
        for (int kx = 0; kx < 3; ++kx) {
            const int dy = 2 * (ky - 1), dx = 2 * (kx - 1);
            const float* wp = W + ((long long)(m0 + mrow) * 128 + 2 * khalf) * 9 + ky * 3 + kx;
            const float* ip = inb + (long long)(2 * khalf) * PCH
                            + (long long)(y + 2 + dy) * 68 + (mrow + 2 + dx);
#pragma unroll 2
            for (int k0 = 0; k0 < 128; k0 += 4) {
                v2f a;
                a.x = wp[k0 * 9];
                a.y = wp[k0 * 9 + 9];
#pragma unroll
                for (int t = 0; t < 4; ++t) {
                    v2f bf;
                    bf.x = ip[(long long)k0 * PCH + t * 16];
                    bf.y = ip[(long long)k0 * PCH + PCH + t * 16];
                    acc[t] = __builtin_amdgcn_wmma_f32_16x16x4_f32(
                        false, a, false, bf, (short)0, acc[t], false, false);
                }
            }
        }
    }
#pragma unroll
    for (int t = 0; t < 4; ++t) {
        const int n = n0 + t * 16 + mrow;
#pragma unroll
        for (int r = 0; r < 8; ++r) {
            const int m = m0 + r + 8 * khalf;
            float v = fmaxf(acc[t][r] + bias[m], 0.0f);
            outb[(long long)m * 4096 + n] = v;
        }
    }
}

// ---------------------------------------------------------------------------
// Exact k-th largest of |D| via 4-pass MSB radix select on abs bit patterns
// (integer order == float order for non-negative floats). 4 selections done
// simultaneously. Fixed launch sequence -> graph-capture safe.
// ---------------------------------------------------------------------------
__global__ void sel_init(unsigned* prefix, unsigned* krem, unsigned* hist, float* thr) {
    const int i = blockIdx.x * blockDim.x + threadIdx.x;
    if (i < 4) {
        prefix[i] = 0u;
        krem[i]   = (unsigned)(TOTAL >> (2 * (i + 1)));   // total/4,/16,/64,/256
        thr[i]    = 0.0f;
    }
    if (i < 1024) hist[i] = 0u;
}

__global__ void __launch_bounds__(256)
radix_hist(const float* __restrict__ D, unsigned* __restrict__ hist,
           const unsigned* __restrict__ prefix, int pass)
{
    __shared__ unsigned lh[1024];
    for (int i = threadIdx.x; i < 1024; i += 256) lh[i] = 0u;
    __syncthreads();
    const int shift = 24 - 8 * pass;
    const unsigned highmask = (pass == 0) ? 0u : (0xFFFFFFFFu << (shift + 8));
    const unsigned p0 = prefix[0], p1 = prefix[1], p2 = prefix[2], p3 = prefix[3];
    const long long stride = (long long)gridDim.x * 256;
    for (long long i = (long long)blockIdx.x * 256 + threadIdx.x; i < TOTAL; i += stride) {
        const unsigned u   = __float_as_uint(D[i]) & 0x7FFFFFFFu;
        const unsigned hi  = u & highmask;
        const unsigned dig = (u >> shift) & 255u;
        if (hi == p0) atomicAdd(&lh[dig], 1u);
        if (hi == p1) atomicAdd(&lh[256 + dig], 1u);
        if (hi == p2) atomicAdd(&lh[512 + dig], 1u);
        if (hi == p3) atomicAdd(&lh[768 + dig], 1u);
    }
    __syncthreads();
    for (int i = threadIdx.x; i < 1024; i += 256) {
        const unsigned c = lh[i];
        if (c) atomicAdd(&hist[i], c);
    }
}

__global__ void radix_scan(unsigned* hist, unsigned* prefix, unsigned* krem,
                           float* thr, int pass)
{
    if (threadIdx.x == 0) {
        const int shift = 24 - 8 * pass;
        for (int s = 0; s < 4; ++s) {
            unsigned* h = hist + s * 256;
            const unsigned k = krem[s];
            unsigned cum = 0;
            int d = 0;
            for (int dig = 255; dig >= 0; --dig) {
                const unsigned c = h[dig];
                if (cum + c >= k) { d = dig; break; }
                cum += c;
            }
            prefix[s] |= ((unsigned)d) << shift;
            krem[s] = k - cum;
            if (pass == 3) thr[s] = __uint_as_float(prefix[s]);
        }
    }
    __syncthreads();
    for (int i = threadIdx.x; i < 1024; i += blockDim.x) hist[i] = 0u;
}

// ---------------------------------------------------------------------------
// Channel softmax over 512 + residual: out = x * softmax_c(outs) + x.
// One wave per (b, m, n) position; wave32 __shfl_xor reductions.
// ---------------------------------------------------------------------------
__global__ void __launch_bounds__(256)
softmax_residual(const float* __restrict__ x, const float* __restrict__ outs,
                 float* __restrict__ out)
{
    const int lane = threadIdx.x & 31;
    const int wave = threadIdx.x >> 5;
    const int pos  = blockIdx.x * 8 + wave;       // 0..32767
    const int b = pos >> 12;
    const int p = pos & 4095;
    const float* oc = outs + (long long)b * 2097152 + p;
    const float* xc = x    + (long long)b * 2097152 + p;
    float*       yc = out  + (long long)b * 2097152 + p;

    float mx = -3.4e38f;
    for (int c = lane; c < 512; c += 32) mx = fmaxf(mx, oc[(long long)c * 4096]);
    for (int o = 16; o > 0; o >>= 1) mx = fmaxf(mx, __shfl_xor(mx, o, 32));

    float sum = 0.0f;
    for (int c = lane; c < 512; c += 32) sum += expf(oc[(long long)c * 4096] - mx);
    for (int o = 16; o > 0; o >>= 1) sum += __shfl_xor(sum, o, 32);
    const float inv = 1.0f / sum;

    for (int c = lane; c < 512; c += 32) {
        const float w  = expf(oc[(long long)c * 4096] - mx) * inv;
        const float xv = xc[(long long)c * 4096];
        yc[(long long)c * 4096] = xv * w + xv;
    }
}

// ---------------------------------------------------------------------------
// Host launcher
// ---------------------------------------------------------------------------
extern "C" void kernel_launch(void* const* d_in, const int* in_sizes, int n_in,
                              void* d_out, int out_size, void* d_ws, size_t ws_size,
                              hipStream_t stream)
{
    const float* x  = (const float*)d_in[0];   // (8,512,64,64)
    const float* W1 = (const float*)d_in[1];   // (4,128,512,1,1)
    const float* b1 = (const float*)d_in[2];   // (4,128)
    const float* W2 = (const float*)d_in[3];   // (4,128,128,3,3)
    const float* b2 = (const float*)d_in[4];   // (4,128)
    const float* W3 = (const float*)d_in[5];   // (4,128,128,1,1)
    const float* b3 = (const float*)d_in[6];   // (4,128)
    float* out = (float*)d_out;

    char* ws = (char*)d_ws;
    size_t off = 0;
    auto alloc = [&](size_t elems) -> float* {
        float* p = (float*)(ws + off);
        off += ((elems * 4 + 255) & ~(size_t)255);
        return p;
    };
    float* Ac   = alloc(4096);              // 64x64 DCT basis (M == N)
    float* Ak   = alloc(262144);            // 512x512 DCT basis
    float* D    = alloc(TOTAL);             // DCT coefficients
    float* T1   = alloc(TOTAL);
    float* T2   = alloc(TOTAL);
    float* OUTS = alloc(TOTAL);             // concat of 4 branch outputs
    float* H1   = alloc((size_t)8 * PB);    // padded (8,128,68,68)
    float* H2   = alloc(4194304);           // (8,128,64,64)
    unsigned* prefix = (unsigned*)alloc(4);
    unsigned* krem   = (unsigned*)alloc(4);
    float*    thr    = alloc(4);
    unsigned* hist   = (unsigned*)alloc(1024);

    const dim3 gblk(128);

    gen_dct<<<1041, 256, 0, stream>>>(Ac, Ak);
    zero_buf<<<4096, 256, 0, stream>>>(H1, (long long)8 * PB);  // halo stays 0 all launch

    // Forward DCT along N: per (b,c), T1[m,k'] = sum_n x[m,n] * Ac[k',n]
    wmma_gemm<4, 64, 1, 1, 64, 64, 1, 64, false, false, false, false>
        <<<dim3(1, 1, 4096), gblk, 0, stream>>>(x, 4096, Ac, 0, T1, 4096, nullptr, nullptr);
    // Forward DCT along M: T2 = Ac * T1 (per b,c)
    wmma_gemm<4, 64, 1, 64, 1, 64, 1, 64, false, false, false, false>
        <<<dim3(1, 1, 4096), gblk, 0, stream>>>(Ac, 0, T1, 4096, T2, 4096, nullptr, nullptr);
    // Forward DCT along C: per b, D = Ak(512x512) * T2(512x4096)
    wmma_gemm<8, 512, 1, 4096, 1, 4096, 1, 512, false, false, false, false>
        <<<dim3(8, 32, 8), gblk, 0, stream>>>(Ak, 0, T2, 2097152, D, 2097152, nullptr, nullptr);

    // Exact top-k thresholds (k-th largest |D|) via 4-pass radix select
    sel_init<<<1, 1024, 0, stream>>>(prefix, krem, hist, thr);
    for (int pass = 0; pass < 4; ++pass) {
        radix_hist<<<2048, 256, 0, stream>>>(D, hist, prefix, pass);
        radix_scan<<<1, 256, 0, stream>>>(hist, prefix, krem, thr, pass);
    }

    for (int i = 0; i < 4; ++i) {
        // Inverse along C with fused mask: T1 = Ak^T * where(|D|>=thr_i, D, 0)
        wmma_gemm<8, 1, 512, 4096, 1, 4096, 1, 512, true, false, false, false>
            <<<dim3(8, 32, 8), gblk, 0, stream>>>(Ak, 0, D, 2097152, T1, 2097152, thr + i, nullptr);
        // Inverse along M: T2 = Ac^T * T1 (per b,c)
        wmma_gemm<4, 1, 64, 64, 1, 64, 1, 64, false, false, false, false>
            <<<dim3(1, 1, 4096), gblk, 0, stream>>>(Ac, 0, T1, 4096, T2, 4096, nullptr, nullptr);
        // Inverse along N: T1 = T2 * Ac
        wmma_gemm<4, 64, 1, 64, 1, 64, 1, 64, false, false, false, false>
            <<<dim3(1, 1, 4096), gblk, 0, stream>>>(T2, 4096, Ac, 0, T1, 4096, nullptr, nullptr);
        // 1x1 conv 512->128 + bias + ReLU into padded H1 (halo pre-zeroed)
        wmma_gemm<8, 512, 1, 4096, 1, 4096, 1, 512, false, true, true, true>
            <<<dim3(2, 32, 8), gblk, 0, stream>>>(W1 + (long long)i * 65536, 0,
                                                  T1, 2097152,
                                                  H1, PB, nullptr, b1 + i * 128);
        // 3x3 dil-2 conv 128->128 + bias + ReLU (padded input, no masks)
        conv3x3_dil2_wmma<<<dim3(2, 64, 8), gblk, 0, stream>>>(
            H1, W2 + (long long)i * 147456, b2 + i * 128, H2);
        // 1x1 conv 128->128 + bias + ReLU into channel block i of OUTS
        wmma_gemm<8, 128, 1, 4096, 1, 4096, 1, 128, false, true, true, false>
            <<<dim3(2, 32, 8), gblk, 0, stream>>>(W3 + (long long)i * 16384, 0,
                                                  H2, 524288,
                                                  OUTS + (long long)i * 524288, 2097152,
                                                  nullptr, b3 + i * 128);
    }

    // Softmax over 512 channels + residual
    softmax_residual<<<4096, 256, 0, stream>>>(x, OUTS, out);
}